// RN_B_15470472200840
// MI455X (gfx1250) — compile-verified
//
#include <hip/hip_runtime.h>

// -------------------------------------------------------------------------
// Region Normalization for MI455X (gfx1250, wave32).
//
//   Stage 1 (rn_stats):  per-(batch, 16-channel group, 1024-pixel tile) block
//                        computes {Σx·m, Σx, Σx²·m, Σx²} for 16 channels with
//                        V_WMMA_F32_16X16X4_F32:  A = [mask; ones; 0...] (16x4),
//                        B = x-tile (4 pixels x 16 channels)  =>  D row0 = Σm·x,
//                        D row1 = Σx per channel, f32-exact. A second WMMA on
//                        x² gives the quadratic sums. Tile fill uses async
//                        global->LDS (ASYNCcnt) when the builtin is available.
//   Stage 2 (rn_coeffs): single block, fixed-order reduce -> closed-form
//                        per-channel affine {A1,B1,A0,B0}.
//   Stage 3 (rn_apply):  out = m ? A1*x+B1 : A0*x+B0, float4 NT-streamed.
//
// Bandwidth-bound: ~0.77 GB traffic -> ~33us at 23.3 TB/s. All reductions are
// fixed-order (no float atomics) so graph replays revalidate identically.
// -------------------------------------------------------------------------

typedef float v2f __attribute__((ext_vector_type(2)));
typedef float v4f __attribute__((ext_vector_type(4)));
typedef float v8f __attribute__((ext_vector_type(8)));
typedef int   gv4i __attribute__((vector_size(4 * sizeof(int))));   // builtin param type
typedef __attribute__((address_space(3))) gv4i* lds_v4i_ptr;        // builtin LDS param

#define NCH      128
#define HW       65536            // 256*256
#define NBATCH   8
#define TILE     1024             // pixels per stage-1 tile
#define XSTRIDE  1028             // TILE+4: stride%64==4 -> conflict-free B reads
#define NTILES   (HW / TILE)      // 64
#define NBLK     (NBATCH * 8 * NTILES)  // 4096 stage-1 blocks
#define NTOT     524288.0f        // B*H*W

#if defined(__AMDGCN__) && __has_builtin(__builtin_amdgcn_global_load_async_to_lds_b128)
#define RN_ASYNC_LDS 1
#endif

// ---------------------------------------------------------------- stage 1 --
__global__ __launch_bounds__(256)
void rn_stats(const float* __restrict__ x, const float* __restrict__ mask,
              float* __restrict__ partials, float* __restrict__ maskparts) {
  extern __shared__ float smem[];          // 16*XSTRIDE + TILE floats
  float* xs = smem;                        // [16][XSTRIDE] x-tile
  float* ms = smem + 16 * XSTRIDE;         // [TILE] mask tile

  const int t    = threadIdx.x;
  const int tile = blockIdx.x;             // 0..63
  const int g    = blockIdx.y;             // channel group 0..7 (16 ch each)
  const int b    = blockIdx.z;             // batch 0..7

  const int base = (b * NCH + g * 16) * HW + tile * TILE;
#if RN_ASYNC_LDS
  // ---- async global->LDS (no VGPR round-trip, tracked by ASYNCcnt) ----
  {
    #pragma unroll
    for (int r = 0; r < 16; ++r) {
      __builtin_amdgcn_global_load_async_to_lds_b128(
          (gv4i*)(x + base + r * HW + t * 4),
          (lds_v4i_ptr)(xs + r * XSTRIDE + t * 4),
          0, /*cpol: TH_NT*/ 1);
    }
    __builtin_amdgcn_global_load_async_to_lds_b128(
        (gv4i*)(mask + b * HW + tile * TILE + t * 4),
        (lds_v4i_ptr)(ms + t * 4), 0, 0);
#if __has_builtin(__builtin_amdgcn_s_wait_asynccnt)
    __builtin_amdgcn_s_wait_asynccnt(0);
#else
    asm volatile("s_wait_asynccnt 0x0" ::: "memory");
#endif
  }
#else
  // ---- fallback: coalesced NT load + LDS store ----
  {
    #pragma unroll 4
    for (int r = 0; r < 16; ++r) {
      v4f v = __builtin_nontemporal_load((const v4f*)(x + base + r * HW + t * 4));
      *(v4f*)(xs + r * XSTRIDE + t * 4) = v;
    }
    v4f mv = *(const v4f*)(mask + b * HW + tile * TILE + t * 4);
    *(v4f*)(ms + t * 4) = mv;
  }
#endif
  __syncthreads();

  const int lane = t & 31;
  const int wave = t >> 5;
  const int ch   = lane & 15;              // A row / B column owned by lane
  const int hh   = lane >> 4;              // 0: K=0,1   1: K=2,3
  // A row0 = mask, row1 = ones, rows 2..15 = 0  (unconditional fma, no exec games)
  const float c0 = (ch == 0) ? 1.0f : 0.0f;
  const float c1 = (ch == 1) ? 1.0f : 0.0f;

  const float* xp = xs + ch * XSTRIDE + 2 * hh;
  const float* mp = ms + 2 * hh;

  v8f dS = {0, 0, 0, 0, 0, 0, 0, 0};       // row0: Σm·x   row1: Σx (per channel)
  v8f dQ = {0, 0, 0, 0, 0, 0, 0, 0};       // row0: Σm·x²  row1: Σx²

  const int q0 = wave * (TILE / 8);        // 128 pixels per wave
  for (int j = 0; j < TILE / 8; j += 4) {
    v2f xv = *(const v2f*)(xp + q0 + j);   // B: 4 pixels x 16 channels
    v2f m2 = *(const v2f*)(mp + q0 + j);   // LDS broadcast (2 addresses)
    v2f av;                                // A: [m; 1; 0...]
    av.x = fmaf(m2.x, c0, c1);
    av.y = fmaf(m2.y, c0, c1);
    dS = __builtin_amdgcn_wmma_f32_16x16x4_f32(false, av, false, xv,
                                               (short)0, dS, false, false);
    v2f xq = xv * xv;
    dQ = __builtin_amdgcn_wmma_f32_16x16x4_f32(false, av, false, xq,
                                               (short)0, dQ, false, false);
  }

  __syncthreads();                         // xs reads done; reuse as scratch
  float* red = smem;                       // [8 waves][16 ch][4 stats]
  if (lane < 16) {                         // D rows 0,1 live in lanes 0-15
    v4f r4 = {dS[0], dS[1], dQ[0], dQ[1]};
    *(v4f*)(red + (wave * 16 + lane) * 4) = r4;
  }
  __syncthreads();

  if (t < 64) {                            // c = t>>2, s = t&3 (Σxm,Σx,Σx²m,Σx²)
    const int c = t >> 2, s = t & 3;
    float sum = 0.0f;
    #pragma unroll
    for (int w = 0; w < 8; ++w) sum += red[(w * 16 + c) * 4 + s];
    const int bid = (b * 8 + g) * NTILES + tile;
    partials[bid * 64 + t] = sum;
  }

  // mask tile sum (only one channel-group needs it; branch is block-uniform)
  if (g == 0) {
    const float m4 = ms[t] + ms[t + 256] + ms[t + 512] + ms[t + 768];
    red[4096 + t] = m4;                    // disjoint from red reads above
    __syncthreads();
    if (t == 0) {
      float s0 = 0.0f;
      for (int k = 0; k < 256; ++k) s0 += red[4096 + k];
      maskparts[b * NTILES + tile] = s0;
    }
  }
}

// ---------------------------------------------------------------- stage 2 --
__global__ __launch_bounds__(640)
void rn_coeffs(const float* __restrict__ partials,
               const float* __restrict__ maskparts,
               const float* __restrict__ fg_gamma, const float* __restrict__ fg_beta,
               const float* __restrict__ bg_gamma, const float* __restrict__ bg_beta,
               float* __restrict__ coeffs) {
  __shared__ float cs[513];
  const int t = threadIdx.x;
  if (t < 512) {                           // channel c = t>>2, stat s = t&3
    const int c = t >> 2, s = t & 3;
    const int g = c >> 4, i = c & 15;
    float sum = 0.0f;
    for (int b = 0; b < NBATCH; ++b)
      for (int tl = 0; tl < NTILES; ++tl)
        sum += partials[((b * 8 + g) * NTILES + tl) * 64 + i * 4 + s];
    cs[t] = sum;
  } else if (t == 512) {
    float nf = 0.0f;
    for (int k = 0; k < NBATCH * NTILES; ++k) nf += maskparts[k];
    cs[512] = nf;
  }
  __syncthreads();
  if (t < NCH) {
    const float S1 = cs[t * 4 + 0], T1 = cs[t * 4 + 1];
    const float S2 = cs[t * 4 + 2], T2 = cs[t * 4 + 3];
    const float Nf = cs[512];
    const float Nb = NTOT - Nf;
    const float Nf1 = (Nf == 0.0f) ? 1.0f : Nf;    // reference's where(Sr==0,1,Sr)
    const float Nb1 = (Nb == 0.0f) ? 1.0f : Nb;
    // foreground RN: fill bg with mu_f, BN, scale by sqrt(Nf/N)
    const float mu_f   = S1 / Nf1;
    const float mean_f = (S1 + Nb * mu_f) / NTOT;
    const float var_f  = (S2 + Nb * mu_f * mu_f) / NTOT - mean_f * mean_f;
    const float inv_f  = rsqrtf(var_f + 1e-5f);
    const float sc_f   = sqrtf(Nf1 / NTOT);
    // background RN
    const float mu_b   = (T1 - S1) / Nb1;
    const float mean_b = ((T1 - S1) + Nf * mu_b) / NTOT;
    const float var_b  = ((T2 - S2) + Nf * mu_b * mu_b) / NTOT - mean_b * mean_b;
    const float inv_b  = rsqrtf(var_b + 1e-5f);
    const float sc_b   = sqrtf(Nb1 / NTOT);
    const float af = sc_f * inv_f, ab = sc_b * inv_b;
    const float gf = 1.0f + fg_gamma[t], gb = 1.0f + bg_gamma[t];
    const float bsum = fg_beta[t] + bg_beta[t];
    // m=1: rn = af*(x-mean_f) + ab*(mu_b-mean_b);  m=0: symmetric
    const float A1 = af * gf;
    const float B1 = (ab * (mu_b - mean_b) - af * mean_f) * gf + bsum;
    const float A0 = ab * gb;
    const float B0 = (af * (mu_f - mean_f) - ab * mean_b) * gb + bsum;
    *(v4f*)(coeffs + t * 4) = (v4f){A1, B1, A0, B0};
  }
}

// ---------------------------------------------------------------- stage 3 --
__global__ __launch_bounds__(256)
void rn_apply(const float* __restrict__ x, const float* __restrict__ mask,
              const float* __restrict__ coeffs, float* __restrict__ out) {
  const int c = blockIdx.y, b = blockIdx.z;
  const v4f k = *(const v4f*)(coeffs + c * 4);       // {A1,B1,A0,B0}
  const int p4 = blockIdx.x * 256 + threadIdx.x;     // float4 idx in plane
  const int pb = (b * NCH + c) * (HW / 4) + p4;
  const v4f xv = *(const v4f*)(x + (size_t)pb * 4);
  const v4f mv = *(const v4f*)(mask + ((size_t)b * (HW / 4) + p4) * 4); // hot in L2
  v4f o;
  #pragma unroll
  for (int e = 0; e < 4; ++e) {
    const float m = mv[e];
    const float a = k[2] + (k[0] - k[2]) * m;        // A0 + (A1-A0)*m, m in {0,1}
    const float bc = k[3] + (k[1] - k[3]) * m;       // B0 + (B1-B0)*m
    o[e] = a * xv[e] + bc;
  }
  __builtin_nontemporal_store(o, (v4f*)(out + (size_t)pb * 4));  // never reread
}

// ------------------------------------------------------------------ launch --
extern "C" void kernel_launch(void* const* d_in, const int* in_sizes, int n_in,
                              void* d_out, int out_size, void* d_ws, size_t ws_size,
                              hipStream_t stream) {
  (void)in_sizes; (void)n_in; (void)out_size; (void)ws_size;
  const float* x        = (const float*)d_in[0];
  const float* mask     = (const float*)d_in[1];
  const float* fg_gamma = (const float*)d_in[2];
  const float* fg_beta  = (const float*)d_in[3];
  const float* bg_gamma = (const float*)d_in[4];
  const float* bg_beta  = (const float*)d_in[5];
  float* out = (float*)d_out;

  float* ws        = (float*)d_ws;
  float* partials  = ws;                         // NBLK*64 floats (1 MB)
  float* maskparts = ws + (size_t)NBLK * 64;     // 512 floats
  float* coeffs    = maskparts + 512;            // 512 floats

  const size_t lds1 = (16 * XSTRIDE + TILE) * sizeof(float);  // ~68 KB of 320 KB
  hipLaunchKernelGGL(rn_stats, dim3(NTILES, 8, NBATCH), dim3(256), lds1, stream,
                     x, mask, partials, maskparts);
  hipLaunchKernelGGL(rn_coeffs, dim3(1), dim3(640), 0, stream,
                     partials, maskparts, fg_gamma, fg_beta, bg_gamma, bg_beta, coeffs);
  hipLaunchKernelGGL(rn_apply, dim3(HW / 1024, NCH, NBATCH), dim3(256), 0, stream,
                     x, mask, coeffs, out);
}